// RTUlayer_80650895884538
// MI455X (gfx1250) — compile-verified
//
#include <hip/hip_runtime.h>

typedef __attribute__((ext_vector_type(2))) float v2f;
typedef __attribute__((ext_vector_type(8))) float v8f;

#define T_STEPS 1024
#define B_SZ    64
#define DIN     128
#define DRECN   256
#define NCOMB   (2*DRECN)            // 512
#define M1      (T_STEPS*B_SZ)       // 65536 rows for u-GEMM
#define M3      (B_SZ*DIN)           // 8192 rows for A-GEMM

// float offsets inside d_out (return order: y, h1,h2,e1n,e2n,e1t,e2t, A1,A2,C1,C2)
#define Y_ELEMS  (T_STEPS*B_SZ*NCOMB)    // 33554432
#define CARRY2D  (B_SZ*DRECN)            // 16384
#define OFF_H1   (Y_ELEMS)
#define OFF_A    (OFF_H1 + 6*CARRY2D)
#define A_ELEMS  (B_SZ*DIN*DRECN)        // 2097152

// ---------------------------------------------------------------------------
// Kernel 0: rotation-power table K[t][0:256] = Re(r^(T-1-t)), K[t][256:512] = Im
// r_j = g_j (cos th_j + i sin th_j), g = sigmoid(nu_log)
// ---------------------------------------------------------------------------
__global__ void rtu_ktable(const float* __restrict__ nu_log,
                           const float* __restrict__ theta,
                           float* __restrict__ K) {
  int tid = blockIdx.x * blockDim.x + threadIdx.x;   // T*DRECN threads
  int t = tid >> 8;
  int j = tid & 255;
  float g  = 1.0f / (1.0f + expf(-nu_log[j]));
  float n  = (float)(T_STEPS - 1 - t);
  float mag = expf(n * logf(g));                      // g^n  (g in (0,1))
  float ang = n * theta[j];
  K[t*NCOMB + j]         = mag * cosf(ang);
  K[t*NCOMB + DRECN + j] = mag * sinf(ang);
}

// ---------------------------------------------------------------------------
// Kernel 1: u = x @ [w1 | w2], written into the y_seq region of d_out.
// One wave computes a 16(M) x 64(N) strip; block = 8 waves = full 512 cols.
// ---------------------------------------------------------------------------
__global__ void rtu_gemm_u(const float* __restrict__ x,
                           const float* __restrict__ w1,
                           const float* __restrict__ w2,
                           float* __restrict__ y) {
  const int lane  = threadIdx.x & 31;
  const int wave  = threadIdx.x >> 5;        // n-strip 0..7
  const int mbase = blockIdx.x * 16;         // [0, 65536)
  const int nbase = wave * 64;               // [0, 512)
  const float* W  = (nbase < DRECN) ? w1 : w2;
  const int wcol  = nbase & (DRECN - 1);
  const int mrow  = mbase + (lane & 15);
  const int khalf = (lane >> 4) * 2;         // 0 or 2
  const int ncol  = lane & 15;

  v8f acc0 = {}, acc1 = {}, acc2 = {}, acc3 = {};
  for (int k = 0; k < DIN; k += 4) {
    // A (16x4): lane -> (m = lane%16, k = k + khalf + v), contiguous pair
    v2f a = *(const v2f*)(x + mrow * DIN + k + khalf);
    // B (4x16): lane -> (k = k + khalf + v, n = nbase + sub*16 + lane%16)
    const float* wp = W + (k + khalf) * DRECN + wcol + ncol;
    v2f b0, b1, b2, b3;
    b0[0] = wp[0];  b0[1] = wp[DRECN];
    b1[0] = wp[16]; b1[1] = wp[DRECN + 16];
    b2[0] = wp[32]; b2[1] = wp[DRECN + 32];
    b3[0] = wp[48]; b3[1] = wp[DRECN + 48];
    acc0 = __builtin_amdgcn_wmma_f32_16x16x4_f32(false, a, false, b0, (short)0, acc0, false, false);
    acc1 = __builtin_amdgcn_wmma_f32_16x16x4_f32(false, a, false, b1, (short)0, acc1, false, false);
    acc2 = __builtin_amdgcn_wmma_f32_16x16x4_f32(false, a, false, b2, (short)0, acc2, false, false);
    acc3 = __builtin_amdgcn_wmma_f32_16x16x4_f32(false, a, false, b3, (short)0, acc3, false, false);
  }
  // C/D layout: VGPR r, lane -> (M = r + 8*(lane/16), N = lane%16)
  float* yo = y + (size_t)(mbase + 8 * (lane >> 4)) * NCOMB + nbase + ncol;
#pragma unroll
  for (int r = 0; r < 8; ++r) {
    yo[r * NCOMB + 0]  = acc0[r];
    yo[r * NCOMB + 16] = acc1[r];
    yo[r * NCOMB + 32] = acc2[r];
    yo[r * NCOMB + 48] = acc3[r];
  }
}

// ---------------------------------------------------------------------------
// Kernel 2: serial scan over T. One thread per (b,j) channel keeps complex
// h, e_nu, e_th in registers; reads u from y region, rewrites it with h.
// ---------------------------------------------------------------------------
__global__ void rtu_scan(const float* __restrict__ nu_log,
                         const float* __restrict__ theta,
                         float* __restrict__ out) {
  int ch = blockIdx.x * blockDim.x + threadIdx.x;   // 16384 channels
  int b = ch >> 8, j = ch & 255;
  float g   = 1.0f / (1.0f + expf(-nu_log[j]));
  float cth = cosf(theta[j]), sth = sinf(theta[j]);
  float c = g * cth, s = g * sth;
  float dg = g * (1.0f - g);
  float dcn = dg * cth, dsn = dg * sth;     // d c/d nu, d s/d nu
  float dct = -g * sth, dst = g * cth;      // d c/d th, d s/d th

  float h1 = 0.f, h2 = 0.f, e1n = 0.f, e2n = 0.f, e1t = 0.f, e2t = 0.f;
  float* yp = out + b * NCOMB + j;
  const int tstride = B_SZ * NCOMB;         // 32768 floats per step
  for (int t = 0; t < T_STEPS; ++t) {
    float u1 = yp[0];
    float u2 = yp[DRECN];
    __builtin_prefetch(yp + tstride, 0, 0);          // global_prefetch_b8
    // eligibility traces use pre-update h
    float t1n = c * e1n - s * e2n + dcn * h1 - dsn * h2;
    float t2n = s * e1n + c * e2n + dsn * h1 + dcn * h2;
    float t1t = c * e1t - s * e2t + dct * h1 - dst * h2;
    float t2t = s * e1t + c * e2t + dst * h1 + dct * h2;
    float nh1 = c * h1 - s * h2 + u1;
    float nh2 = s * h1 + c * h2 + u2;
    yp[0]     = nh1;
    yp[DRECN] = nh2;
    h1 = nh1; h2 = nh2; e1n = t1n; e2n = t2n; e1t = t1t; e2t = t2t;
    yp += tstride;
  }
  out[OFF_H1 + 0 * CARRY2D + ch] = h1;
  out[OFF_H1 + 1 * CARRY2D + ch] = h2;
  out[OFF_H1 + 2 * CARRY2D + ch] = e1n;
  out[OFF_H1 + 3 * CARRY2D + ch] = e2n;
  out[OFF_H1 + 4 * CARRY2D + ch] = e1t;
  out[OFF_H1 + 5 * CARRY2D + ch] = e2t;
}

// ---------------------------------------------------------------------------
// Kernel 3: A(T) = sum_t r^(T-1-t) x[t]  as GEMM  x^T (8192x1024) @ K (1024x512).
// Columns 0..255 -> A1 (and C2 = A1); columns 256..511 -> A2 (and C1 = -A2).
// ---------------------------------------------------------------------------
__global__ void rtu_gemm_A(const float* __restrict__ x,   // [T][8192]
                           const float* __restrict__ K,   // [T][512]
                           float* __restrict__ out) {
  const int lane  = threadIdx.x & 31;
  const int wave  = threadIdx.x >> 5;
  const int mbase = blockIdx.x * 16;         // bi tile, [0, 8192)
  const int nbase = wave * 64;               // [0, 512)
  const int mrow  = mbase + (lane & 15);
  const int khalf = (lane >> 4) * 2;
  const int ncol  = lane & 15;

  v8f acc0 = {}, acc1 = {}, acc2 = {}, acc3 = {};
  for (int k = 0; k < T_STEPS; k += 4) {
    v2f a;                                    // x^T tile: (bi, t) = x[t*8192+bi]
    a[0] = x[(k + khalf) * M3 + mrow];
    a[1] = x[(k + khalf + 1) * M3 + mrow];
    const float* kp = K + (k + khalf) * NCOMB + nbase + ncol;
    v2f b0, b1, b2, b3;
    b0[0] = kp[0];  b0[1] = kp[NCOMB];
    b1[0] = kp[16]; b1[1] = kp[NCOMB + 16];
    b2[0] = kp[32]; b2[1] = kp[NCOMB + 32];
    b3[0] = kp[48]; b3[1] = kp[NCOMB + 48];
    acc0 = __builtin_amdgcn_wmma_f32_16x16x4_f32(false, a, false, b0, (short)0, acc0, false, false);
    acc1 = __builtin_amdgcn_wmma_f32_16x16x4_f32(false, a, false, b1, (short)0, acc1, false, false);
    acc2 = __builtin_amdgcn_wmma_f32_16x16x4_f32(false, a, false, b2, (short)0, acc2, false, false);
    acc3 = __builtin_amdgcn_wmma_f32_16x16x4_f32(false, a, false, b3, (short)0, acc3, false, false);
  }

  const int row0 = mbase + 8 * (lane >> 4);
  const int j0   = (nbase & (DRECN - 1)) + ncol;
  if (nbase < DRECN) {               // these columns are A1; C2 = A1
    float* pA1 = out + OFF_A + 0 * A_ELEMS;
    float* pC2 = out + OFF_A + 3 * A_ELEMS;
#pragma unroll
    for (int r = 0; r < 8; ++r) {
      int base = (row0 + r) * DRECN + j0;
      pA1[base + 0]  = acc0[r];  pC2[base + 0]  = acc0[r];
      pA1[base + 16] = acc1[r];  pC2[base + 16] = acc1[r];
      pA1[base + 32] = acc2[r];  pC2[base + 32] = acc2[r];
      pA1[base + 48] = acc3[r];  pC2[base + 48] = acc3[r];
    }
  } else {                           // these columns are A2; C1 = -A2
    float* pA2 = out + OFF_A + 1 * A_ELEMS;
    float* pC1 = out + OFF_A + 2 * A_ELEMS;
#pragma unroll
    for (int r = 0; r < 8; ++r) {
      int base = (row0 + r) * DRECN + j0;
      pA2[base + 0]  = acc0[r];  pC1[base + 0]  = -acc0[r];
      pA2[base + 16] = acc1[r];  pC1[base + 16] = -acc1[r];
      pA2[base + 32] = acc2[r];  pC1[base + 32] = -acc2[r];
      pA2[base + 48] = acc3[r];  pC1[base + 48] = -acc3[r];
    }
  }
}

// ---------------------------------------------------------------------------
extern "C" void kernel_launch(void* const* d_in, const int* in_sizes, int n_in,
                              void* d_out, int out_size, void* d_ws, size_t ws_size,
                              hipStream_t stream) {
  const float* x      = (const float*)d_in[0];
  const float* w1     = (const float*)d_in[1];
  const float* w2     = (const float*)d_in[2];
  const float* nu_log = (const float*)d_in[3];
  const float* theta  = (const float*)d_in[4];
  float* out = (float*)d_out;
  float* K   = (float*)d_ws;                 // T*512 floats = 2 MB scratch

  rtu_ktable<<<(T_STEPS * DRECN) / 256, 256, 0, stream>>>(nu_log, theta, K);
  rtu_gemm_u<<<M1 / 16, 256, 0, stream>>>(x, w1, w2, out);
  rtu_scan  <<<(B_SZ * DRECN) / 256, 256, 0, stream>>>(nu_log, theta, out);
  rtu_gemm_A<<<M3 / 16, 256, 0, stream>>>(x, K, out);
}